// CAGAT_MinSum_Layer_Lite_88802743812476
// MI455X (gfx1250) — compile-verified
//
#include <hip/hip_runtime.h>
#include <stdint.h>

// ---------------------------------------------------------------------------
// CAGAT MinSum edge kernel for MI455X (gfx1250).
// Memory-bound gather/scatter: stream edges via CDNA5 async-to-LDS DMA
// (double buffered, NT cache hint), gather node features from L2, scatter
// with hardware global_atomic_add_f32 at device scope.
// ---------------------------------------------------------------------------

#define BLOCK_THREADS   256
#define WAVES_PER_BLOCK (BLOCK_THREADS / 32)
#define TILE_EDGES      128                      // edges per wave per iteration
#define TILE_BYTES      (128 * 8 + 128 * 8 + 128 * 4)   // src + dst + mask = 2560
#define WAVE_LDS_BYTES  (2 * TILE_BYTES)                // double buffered = 5120
#define BLOCK_LDS_BYTES (WAVES_PER_BLOCK * WAVE_LDS_BYTES) // 40960 (< 320KB/WGP)

// One lane copies 16B; a wave32 copies 512B per instruction.
__device__ __forceinline__ void async_b128(unsigned lds_off, unsigned long long gaddr) {
  asm volatile("global_load_async_to_lds_b128 %0, %1, off th:TH_LOAD_NT"
               :: "v"(lds_off), "v"(gaddr) : "memory");
}

// Hardware float atomic add, device scope (performed at L2, no CAS loop).
__device__ __forceinline__ void atomic_add_f32(float* p, float v) {
  asm volatile("global_atomic_add_f32 %0, %1, off scope:SCOPE_DEV"
               :: "v"((unsigned long long)(uintptr_t)p), "v"(v) : "memory");
}

// Issue the 5 async copies for one 128-edge tile (1KB src, 1KB dst, 512B mask).
__device__ __forceinline__ void issue_tile(unsigned ldsBuf,
                                           const unsigned long long* __restrict__ src,
                                           const unsigned long long* __restrict__ dst,
                                           const float* __restrict__ mask,
                                           long long base, unsigned lane) {
  unsigned long long gs = (unsigned long long)(uintptr_t)(src  + base);
  unsigned long long gd = (unsigned long long)(uintptr_t)(dst  + base);
  unsigned long long gm = (unsigned long long)(uintptr_t)(mask + base);
  unsigned o = lane * 16u;
  async_b128(ldsBuf +          o, gs +        o);
  async_b128(ldsBuf +  512u +  o, gs + 512u + o);
  async_b128(ldsBuf + 1024u +  o, gd +        o);
  async_b128(ldsBuf + 1536u +  o, gd + 512u + o);
  async_b128(ldsBuf + 2048u +  o, gm +        o);
}

__device__ __forceinline__ float edge_msg(float xs, float xd, float m,
                                          float w0, float w1, float w2,
                                          float bb, float scal, float pen) {
  float raw = fmaf(xs, w0, fmaf(xd, w1, fmaf(m, w2, bb)));
  raw = raw > 0.0f ? raw : 0.01f * raw;           // leaky_relu(0.01)
  raw = fmaf(m, pen, raw);                        // + cycle_mask * cycle_penalty
  float att = __builtin_amdgcn_rcpf(1.0f + __expf(-raw));  // sigmoid via v_exp/v_rcp
  return xs * att * scal;
}

__global__ __launch_bounds__(BLOCK_THREADS)
void cagat_minsum_edge_kernel(const float* __restrict__ nf,
                              const unsigned long long* __restrict__ src,
                              const unsigned long long* __restrict__ dst,
                              const float* __restrict__ mask,
                              const float* __restrict__ W,
                              const float* __restrict__ B,
                              const float* __restrict__ SC,
                              const float* __restrict__ CP,
                              float* __restrict__ out,
                              long long E) {
  __shared__ __attribute__((aligned(16))) unsigned char smem[BLOCK_LDS_BYTES];

  const unsigned lane = threadIdx.x & 31u;
  const unsigned wave = threadIdx.x >> 5;

  // Uniform scalars -> s_load once.
  const float w0 = W[0], w1 = W[1], w2 = W[2];
  const float bb = B[0], scal = SC[0], pen = CP[0];

  const long long nTiles  = E >> 7;                                   // full 128-edge tiles
  const long long wid     = (long long)blockIdx.x * WAVES_PER_BLOCK + wave;
  const long long wstride = (long long)gridDim.x * WAVES_PER_BLOCK;

  // Low 32 bits of a generic LDS pointer == LDS byte offset (hardware dsaddr).
  const unsigned smemBase = (unsigned)(uintptr_t)(void*)smem;
  const unsigned buf0 = smemBase + wave * WAVE_LDS_BYTES;
  const unsigned buf1 = buf0 + TILE_BYTES;
  unsigned char* const bufp0 = smem + wave * WAVE_LDS_BYTES;
  unsigned char* const bufp1 = bufp0 + TILE_BYTES;

  if (wid < nTiles) {
    issue_tile(buf0, src, dst, mask, wid << 7, lane);   // prime buffer 0
    unsigned cur = 0;
    for (long long t = wid; t < nTiles; t += wstride) {
      long long nxt = t + wstride;
      long long pre = (nxt < nTiles) ? nxt : t;         // keep 5 ops in flight (uniform)
      issue_tile(cur ? buf0 : buf1, src, dst, mask, pre << 7, lane);

      // 5 newest (prefetch) may remain outstanding; async loads complete
      // in order, so <=5 means the current tile's 5 copies are done.
      asm volatile("s_wait_asynccnt 0x5" ::: "memory");

      unsigned char* bp = cur ? bufp1 : bufp0;
      const ulonglong2* sv = reinterpret_cast<const ulonglong2*>(bp +          lane * 32u);
      const ulonglong2* dv = reinterpret_cast<const ulonglong2*>(bp + 1024u + lane * 32u);
      const float4*     mv = reinterpret_cast<const float4*>    (bp + 2048u + lane * 16u);
      ulonglong2 s01 = sv[0], s23 = sv[1];              // ds_load_b128 x4
      ulonglong2 d01 = dv[0], d23 = dv[1];
      float4 m4 = *mv;

      // Batch all 8 L2 gathers before any atomic (keeps loads coalesced in flight).
      float xs0 = nf[s01.x], xs1 = nf[s01.y], xs2 = nf[s23.x], xs3 = nf[s23.y];
      float xd0 = nf[d01.x], xd1 = nf[d01.y], xd2 = nf[d23.x], xd3 = nf[d23.y];

      float g0 = edge_msg(xs0, xd0, m4.x, w0, w1, w2, bb, scal, pen);
      float g1 = edge_msg(xs1, xd1, m4.y, w0, w1, w2, bb, scal, pen);
      float g2 = edge_msg(xs2, xd2, m4.z, w0, w1, w2, bb, scal, pen);
      float g3 = edge_msg(xs3, xd3, m4.w, w0, w1, w2, bb, scal, pen);

      atomic_add_f32(out + d01.x, g0);
      atomic_add_f32(out + d01.y, g1);
      atomic_add_f32(out + d23.x, g2);
      atomic_add_f32(out + d23.y, g3);

      cur ^= 1;
    }
  }

  // Tail (E not a multiple of 128): direct loads, one edge per thread.
  long long tail = nTiles << 7;
  long long e = tail + (long long)blockIdx.x * blockDim.x + threadIdx.x;
  if (e < E) {
    unsigned long long s = src[e], d = dst[e];
    float m  = mask[e];
    float xs = nf[s], xd = nf[d];
    atomic_add_f32(out + d, edge_msg(xs, xd, m, w0, w1, w2, bb, scal, pen));
  }
}

extern "C" void kernel_launch(void* const* d_in, const int* in_sizes, int n_in,
                              void* d_out, int out_size, void* d_ws, size_t ws_size,
                              hipStream_t stream) {
  (void)n_in; (void)d_ws; (void)ws_size;
  const float* nf               = (const float*)d_in[0];
  const unsigned long long* ei  = (const unsigned long long*)d_in[1];  // int64 per reference
  const float* mask             = (const float*)d_in[2];
  const float* W                = (const float*)d_in[3];
  const float* B                = (const float*)d_in[4];
  const float* SC               = (const float*)d_in[5];
  const float* CP               = (const float*)d_in[6];
  float* out                    = (float*)d_out;

  const long long E = (long long)in_sizes[2];        // cycle_mask length == n_edges
  const unsigned long long* src = ei;
  const unsigned long long* dst = ei + E;

  // segment_sum accumulates from zero; d_out is poisoned -> zero it each call.
  hipMemsetAsync(d_out, 0, (size_t)out_size * sizeof(float), stream);

  const int blocks = 2048;   // 16384 waves; ~16 tiles/wave; 40KB LDS/block (8 blocks/WGP fit)
  cagat_minsum_edge_kernel<<<blocks, BLOCK_THREADS, 0, stream>>>(
      nf, src, dst, mask, W, B, SC, CP, out, E);
}